// YOLOLoss_37048387895729
// MI455X (gfx1250) — compile-verified
//
#include <hip/hip_runtime.h>
#include <math.h>

typedef float v2f __attribute__((ext_vector_type(2)));
typedef float v8f __attribute__((ext_vector_type(8)));

#define LAMBDA_COORD 5.0f
#define LAMBDA_NOOBJ 0.5f
#define IMGSZ 448.0f
#define CELLSZ 64.0f
#define EPSV 1e-6f

__device__ __forceinline__ float clip448(float v) {
    return fminf(fmaxf(v, 0.0f), IMGSZ);
}
__device__ __forceinline__ float sq(float v) { return v * v; }

// Exact wave32 sum via V_WMMA_F32_16X16X4_F32 (sum == matvec with ones).
// Step 1: D1 = A x ones  -> D1[m][n] = rowsum_m (each lane's partial occupies
//         one (M,K) slot of the 16x4 A; bijection => sum is layout-independent).
// Step 2: four accumulating D2 += ones x B, with B fed pairs of D1 VGPRs
//         (each half-VGPR of B is one full K-row) -> total broadcast to all lanes.
// Requires EXEC all-ones at call site: call only from fully convergent code.
__device__ __forceinline__ float wave32_sum_wmma(float v) {
    v2f a;   a[0] = v;    a[1] = 0.0f;
    v2f one; one[0] = 1.0f; one[1] = 1.0f;
    v8f z = {};
    v8f d1 = __builtin_amdgcn_wmma_f32_16x16x4_f32(false, a, false, one,
                                                   (short)0, z, false, false);
    v8f d2 = {};
    v2f b;
    b[0] = d1[0]; b[1] = d1[1];
    d2 = __builtin_amdgcn_wmma_f32_16x16x4_f32(false, one, false, b,
                                               (short)0, d2, false, false);
    b[0] = d1[2]; b[1] = d1[3];
    d2 = __builtin_amdgcn_wmma_f32_16x16x4_f32(false, one, false, b,
                                               (short)0, d2, false, false);
    b[0] = d1[4]; b[1] = d1[5];
    d2 = __builtin_amdgcn_wmma_f32_16x16x4_f32(false, one, false, b,
                                               (short)0, d2, false, false);
    b[0] = d1[6]; b[1] = d1[7];
    d2 = __builtin_amdgcn_wmma_f32_16x16x4_f32(false, one, false, b,
                                               (short)0, d2, false, false);
    return d2[0];
}

// Stage 1: one thread per grid cell; per-block partial sums to d_ws.
__global__ void __launch_bounds__(256)
yolo_loss_partial(const float* __restrict__ in,   // [B,30,7,7]
                  const float* __restrict__ tg,   // [B,7,7,30]
                  float* __restrict__ partial, int N) {
    int i = blockIdx.x * blockDim.x + threadIdx.x;
    float v = 0.0f;
    if (i < N) {
        int b    = i / 49;
        int cell = i - b * 49;
        int r    = cell / 7;
        int c    = cell - r * 7;

        // input: 30 channels, stride 49 within batch (coalesced across lanes)
        const float* xin = in + (size_t)b * 1470 + (size_t)cell;
        float x[30];
#pragma unroll
        for (int j = 0; j < 30; ++j) x[j] = xin[j * 49];

        // target: 30 contiguous floats, 8B aligned -> float2 loads
        float t[30];
        const float2* tp = reinterpret_cast<const float2*>(tg + (size_t)i * 30);
#pragma unroll
        for (int j = 0; j < 15; ++j) { float2 w = tp[j]; t[2*j] = w.x; t[2*j+1] = w.y; }

        float colf = (float)c * CELLSZ;
        float rowf = (float)r * CELLSZ;

        float iou[2];
#pragma unroll
        for (int bx = 0; bx < 2; ++bx) {
            int o = bx * 5;
            float pcx = fmaf(x[o + 0], CELLSZ, colf);
            float pcy = fmaf(x[o + 1], CELLSZ, rowf);
            float pw  = x[o + 2] * IMGSZ;
            float ph  = x[o + 3] * IMGSZ;
            float px1 = clip448(pcx - 0.5f * pw), px2 = clip448(pcx + 0.5f * pw);
            float py1 = clip448(pcy - 0.5f * ph), py2 = clip448(pcy + 0.5f * ph);

            float tcx = fmaf(t[o + 0], CELLSZ, colf);
            float tcy = fmaf(t[o + 1], CELLSZ, rowf);
            float tw  = t[o + 2] * IMGSZ;
            float th  = t[o + 3] * IMGSZ;
            float tx1 = clip448(tcx - 0.5f * tw), tx2 = clip448(tcx + 0.5f * tw);
            float ty1 = clip448(tcy - 0.5f * th), ty2 = clip448(tcy + 0.5f * th);

            float iw = fmaxf(fminf(px2, tx2) - fmaxf(px1, tx1), EPSV);
            float ih = fmaxf(fminf(py2, ty2) - fmaxf(py1, ty1), EPSV);
            float inter = iw * ih;
            float asum  = (px2 - px1) * (py2 - py1) + (tx2 - tx1) * (ty2 - ty1);
            iou[bx] = inter / (asum - inter);
        }

        bool  m  = iou[0] < iou[1];
        float bi = m ? iou[1] : iou[0];
        float s0 = m ? x[5] : x[0];
        float s1 = m ? x[6] : x[1];
        float s2 = m ? x[7] : x[2];
        float s3 = m ? x[8] : x[3];
        float s4 = m ? x[9] : x[4];

        float has = (t[4] == 1.0f) ? 1.0f : 0.0f;

        float coord = sq(s0 - t[0]) + sq(s1 - t[1]);
        float size  = sq(sqrtf(s2) - sqrtf(t[2])) + sq(sqrtf(s3) - sqrtf(t[3]));
        float conf  = sq(s4 - bi);
        float cls = 0.0f;
#pragma unroll
        for (int j = 10; j < 30; ++j) cls += sq(x[j] - t[j]);
        float noobj = x[4] * x[4] + x[9] * x[9];

        v = has * (LAMBDA_COORD * (coord + size) + conf + cls)
          + (1.0f - has) * (LAMBDA_NOOBJ * noobj);
    }

    float wsum = wave32_sum_wmma(v);   // fully convergent here

    __shared__ float ws[8];
    int lane = threadIdx.x & 31;
    int wid  = threadIdx.x >> 5;
    if (lane == 0) ws[wid] = wsum;
    __syncthreads();
    if (threadIdx.x == 0) {
        float s = 0.0f;
#pragma unroll
        for (int w = 0; w < 8; ++w) s += ws[w];
        partial[blockIdx.x] = s;
    }
}

// Stage 2: single block deterministically folds per-block partials.
__global__ void __launch_bounds__(256)
yolo_loss_finalize(const float* __restrict__ partial, float* __restrict__ out,
                   int nblocks, float inv_b) {
    float v = 0.0f;
    for (int j = threadIdx.x; j < nblocks; j += 256) v += partial[j];

    float wsum = wave32_sum_wmma(v);   // reconverged after uniform-bound loop

    __shared__ float ws[8];
    int lane = threadIdx.x & 31;
    int wid  = threadIdx.x >> 5;
    if (lane == 0) ws[wid] = wsum;
    __syncthreads();
    if (threadIdx.x == 0) {
        float s = 0.0f;
#pragma unroll
        for (int w = 0; w < 8; ++w) s += ws[w];
        out[0] = s * inv_b;
    }
}

extern "C" void kernel_launch(void* const* d_in, const int* in_sizes, int n_in,
                              void* d_out, int out_size, void* d_ws, size_t ws_size,
                              hipStream_t stream) {
    const float* in = (const float*)d_in[0];   // [B,30,7,7]
    const float* tg = (const float*)d_in[1];   // [B,7,7,30]
    float* out = (float*)d_out;
    float* partial = (float*)d_ws;

    int total = in_sizes[0];          // B*30*49
    int N = total / 30;               // B*49 cells = 802816
    int B = total / 1470;             // 16384
    int blocks = (N + 255) / 256;     // 3136 (exact)

    yolo_loss_partial<<<blocks, 256, 0, stream>>>(in, tg, partial, N);
    yolo_loss_finalize<<<1, 256, 0, stream>>>(partial, out, blocks, 1.0f / (float)B);
}